// SimpleDiffPool_38749194945062
// MI455X (gfx1250) — compile-verified
//
#include <hip/hip_runtime.h>

typedef __attribute__((ext_vector_type(2))) float v2f;
typedef __attribute__((ext_vector_type(8))) float v8f;

#define TOTAL  65536      // B*N nodes
#define NPG    1024       // nodes per graph
#define NBATCH 64
#define EMBD   64
#define HIDD   64
#define CDIM   32
#define NCLS   2

// ---------------- WMMA helper (f32, 16x16xK4) ----------------
__device__ __forceinline__ v8f wmma4(v2f a, v2f b, v8f c) {
  // 8 args: (neg_a, A, neg_b, B, c_mod, C, reuse_a, reuse_b)
  return __builtin_amdgcn_wmma_f32_16x16x4_f32(false, a, false, b, (short)0, c,
                                               false, false);
}

// ---------------- degree / norm ----------------
__global__ void k_init_deg(float* deg) {
  int i = blockIdx.x * blockDim.x + threadIdx.x;
  if (i < TOTAL) deg[i] = 1.0f;  // self loop
}
__global__ void k_deg(const int* __restrict__ dst, int E, float* deg) {
  int i = blockIdx.x * blockDim.x + threadIdx.x;
  if (i < E) atomicAdd(&deg[dst[i]], 1.0f);
}
__global__ void k_dinv(const float* __restrict__ deg, float* __restrict__ dinv) {
  int i = blockIdx.x * blockDim.x + threadIdx.x;
  if (i < TOTAL) {
    float d = deg[i];
    dinv[i] = (d > 0.f) ? rsqrtf(d) : 0.f;
  }
}

// ---------------- fused embedding gather + GCN GEMM: h = emb[x_idx] @ W_g ----
// one wave computes 16 rows x 64 cols, K=64 via 16 steps of 16x16x4 WMMA
__global__ void __launch_bounds__(128)
k_embed_gemm(const int* __restrict__ xidx, const float* __restrict__ emb,
             const float* __restrict__ Wg, float* __restrict__ h) {
  int wave    = (int)((blockIdx.x * blockDim.x + threadIdx.x) >> 5);
  int lane    = threadIdx.x & 31;
  int rowbase = wave * 16;
  int M       = lane & 15;
  int koff    = (lane >> 4) * 2;
  int n0      = lane & 15;
  long long ebase = (long long)xidx[rowbase + M] * EMBD;

  v8f c0 = {}, c1 = {}, c2 = {}, c3 = {};
  for (int kk = 0; kk < 16; ++kk) {
    int kb = kk * 4 + koff;
    v2f a;
    a.x = emb[ebase + kb];
    a.y = emb[ebase + kb + 1];
    v2f b;
    b.x = Wg[kb * HIDD + n0];      b.y = Wg[(kb + 1) * HIDD + n0];      c0 = wmma4(a, b, c0);
    b.x = Wg[kb * HIDD + n0 + 16]; b.y = Wg[(kb + 1) * HIDD + n0 + 16]; c1 = wmma4(a, b, c1);
    b.x = Wg[kb * HIDD + n0 + 32]; b.y = Wg[(kb + 1) * HIDD + n0 + 32]; c2 = wmma4(a, b, c2);
    b.x = Wg[kb * HIDD + n0 + 48]; b.y = Wg[(kb + 1) * HIDD + n0 + 48]; c3 = wmma4(a, b, c3);
  }
  int ro = rowbase + 8 * (lane >> 4);
  for (int v = 0; v < 8; ++v) {
    long long rr = (long long)(ro + v) * HIDD;
    h[rr + n0]      = c0[v];
    h[rr + n0 + 16] = c1[v];
    h[rr + n0 + 32] = c2[v];
    h[rr + n0 + 48] = c3[v];
  }
}

// ---------------- edge scatter: z[d] += dinv[s]*dinv[d] * h[s] --------------
__global__ void k_scatter(const int* __restrict__ src, const int* __restrict__ dst,
                          int E, const float* __restrict__ dinv,
                          const float* __restrict__ h, float* __restrict__ z) {
  int g = blockIdx.x * blockDim.x + threadIdx.x;
  int e = g >> 5, lane = g & 31;
  if (e >= E) return;
  int s = src[e], d = dst[e];
  float nm = dinv[s] * dinv[d];
  int c = lane * 2;
  long long sb = (long long)s * HIDD + c;
  long long db = (long long)d * HIDD + c;
  atomicAdd(&z[db],     nm * h[sb]);
  atomicAdd(&z[db + 1], nm * h[sb + 1]);
}

// ---------------- z = relu(z + dinv^2*h + b_g) ------------------------------
__global__ void k_finish_z(float* __restrict__ z, const float* __restrict__ h,
                           const float* __restrict__ dinv, const float* __restrict__ bg) {
  int t = blockIdx.x * blockDim.x + threadIdx.x;
  if (t >= TOTAL * HIDD) return;
  int i = t >> 6, c = t & 63;
  float di = dinv[i];
  float v  = z[t] + di * di * h[t] + bg[c];
  z[t] = v > 0.f ? v : 0.f;
}

// ---------------- assignment MLP + double softmax + entropy ----------------
// per wave: H1 = relu(Z(16x64)@Wa1 + ba1) [to LDS]; L = H1@Wa2 + ba2 (16x32);
// lanes 0..15 do row-wise double softmax, write S, accumulate entropy
__global__ void __launch_bounds__(128)
k_assign(const float* __restrict__ z, const float* __restrict__ Wa1,
         const float* __restrict__ ba1, const float* __restrict__ Wa2,
         const float* __restrict__ ba2, float* __restrict__ S, float* __restrict__ acc) {
  __shared__ float ldsH[4][16 * HIDD];
  __shared__ float ldsS[4][16 * CDIM];
  int wv      = threadIdx.x >> 5;
  int lane    = threadIdx.x & 31;
  int wave    = blockIdx.x * 4 + wv;
  int rowbase = wave * 16;
  int M       = lane & 15;
  int koff    = (lane >> 4) * 2;
  int n0      = lane & 15;

  // GEMM1: 16x64 @ 64x64
  v8f a0 = {}, a1 = {}, a2 = {}, a3 = {};
  const float* Arow = z + (long long)(rowbase + M) * HIDD;
  for (int kk = 0; kk < 16; ++kk) {
    int kb = kk * 4 + koff;
    v2f a; a.x = Arow[kb]; a.y = Arow[kb + 1];
    v2f b;
    b.x = Wa1[kb * HIDD + n0];      b.y = Wa1[(kb + 1) * HIDD + n0];      a0 = wmma4(a, b, a0);
    b.x = Wa1[kb * HIDD + n0 + 16]; b.y = Wa1[(kb + 1) * HIDD + n0 + 16]; a1 = wmma4(a, b, a1);
    b.x = Wa1[kb * HIDD + n0 + 32]; b.y = Wa1[(kb + 1) * HIDD + n0 + 32]; a2 = wmma4(a, b, a2);
    b.x = Wa1[kb * HIDD + n0 + 48]; b.y = Wa1[(kb + 1) * HIDD + n0 + 48]; a3 = wmma4(a, b, a3);
  }
  int ro = 8 * (lane >> 4);
  for (int v = 0; v < 8; ++v) {
    int rr = (ro + v) * HIDD;
    ldsH[wv][rr + n0]      = fmaxf(a0[v] + ba1[n0], 0.f);
    ldsH[wv][rr + n0 + 16] = fmaxf(a1[v] + ba1[n0 + 16], 0.f);
    ldsH[wv][rr + n0 + 32] = fmaxf(a2[v] + ba1[n0 + 32], 0.f);
    ldsH[wv][rr + n0 + 48] = fmaxf(a3[v] + ba1[n0 + 48], 0.f);
  }
  __syncthreads();

  // GEMM2: 16x64 @ 64x32
  v8f s0 = {}, s1 = {};
  for (int kk = 0; kk < 16; ++kk) {
    int kb = kk * 4 + koff;
    v2f a; a.x = ldsH[wv][M * HIDD + kb]; a.y = ldsH[wv][M * HIDD + kb + 1];
    v2f b;
    b.x = Wa2[kb * CDIM + n0];      b.y = Wa2[(kb + 1) * CDIM + n0];      s0 = wmma4(a, b, s0);
    b.x = Wa2[kb * CDIM + n0 + 16]; b.y = Wa2[(kb + 1) * CDIM + n0 + 16]; s1 = wmma4(a, b, s1);
  }
  for (int v = 0; v < 8; ++v) {
    int rr = (ro + v) * CDIM;
    ldsS[wv][rr + n0]      = s0[v] + ba2[n0];
    ldsS[wv][rr + n0 + 16] = s1[v] + ba2[n0 + 16];
  }
  __syncthreads();

  float ent = 0.f;
  if (lane < 16) {
    float* row = &ldsS[wv][lane * CDIM];
    float p[CDIM];
    float m = -1e30f;
    for (int c = 0; c < CDIM; ++c) { p[c] = row[c]; m = fmaxf(m, p[c]); }
    float sm = 0.f;
    for (int c = 0; c < CDIM; ++c) { p[c] = __expf(p[c] - m); sm += p[c]; }
    float inv = 1.f / sm;
    float m2 = -1e30f;
    for (int c = 0; c < CDIM; ++c) { p[c] *= inv; m2 = fmaxf(m2, p[c]); }
    sm = 0.f;
    for (int c = 0; c < CDIM; ++c) { p[c] = __expf(p[c] - m2); sm += p[c]; }
    inv = 1.f / sm;
    long long ob = (long long)(rowbase + lane) * CDIM;
    for (int c = 0; c < CDIM; ++c) {
      float q = p[c] * inv;
      S[ob + c] = q;
      ent -= q * __logf(q + 1e-15f);
    }
  }
  for (int off = 8; off; off >>= 1) ent += __shfl_down(ent, off);
  if (lane == 0) atomicAdd(&acc[2], ent);
}

// ---------------- per-edge loss terms: sum(A^2) and sum(A .* S S^T) ---------
__global__ void k_edge_loss(const int* __restrict__ src, const int* __restrict__ dst,
                            int E, const float* __restrict__ S,
                            unsigned int* __restrict__ Apk, int use_a,
                            float* __restrict__ acc) {
  __shared__ float blk[2];
  if (threadIdx.x == 0) { blk[0] = 0.f; blk[1] = 0.f; }
  __syncthreads();
  int g = blockIdx.x * blockDim.x + threadIdx.x;
  int e = g >> 5, lane = g & 31;
  if (e < E) {
    int s = src[e], d = dst[e];
    float prod = S[(long long)s * CDIM + lane] * S[(long long)d * CDIM + lane];
    for (int off = 16; off; off >>= 1) prod += __shfl_down(prod, off);
    if (lane == 0) {
      atomicAdd(&blk[1], prod);
      float a2c = 1.f;
      if (use_a) {
        unsigned int b    = (unsigned int)(s / NPG);
        unsigned int slot = b * (unsigned int)(NPG * NPG) +
                            (unsigned int)(s % NPG) * NPG + (unsigned int)(d % NPG);
        unsigned int w = slot >> 2, sub = (slot & 3u) * 8u;
        unsigned int old = atomicAdd(&Apk[w], 1u << sub);
        unsigned int oc  = (old >> sub) & 0xFFu;
        a2c = (float)(2u * oc + 1u);
      }
      atomicAdd(&blk[0], a2c);
    }
  }
  __syncthreads();
  if (threadIdx.x == 0) {
    atomicAdd(&acc[0], blk[0]);
    atomicAdd(&acc[1], blk[1]);
  }
}

// ---------------- Gram per graph: sum(SS^T ^2) = ||S^T S||_F^2 --------------
__global__ void __launch_bounds__(256)
k_gram(const float* __restrict__ S, float* __restrict__ acc) {
  __shared__ float tile[128 * CDIM];
  int b = blockIdx.x;
  int t = threadIdx.x;
  int c0 = t >> 5, d0 = t & 31;  // entries (c0 + 8k, d0), k=0..3
  float g0 = 0, g1 = 0, g2 = 0, g3 = 0;
  for (int chunk = 0; chunk < NPG; chunk += 128) {
    const float* base = S + ((long long)b * NPG + chunk) * CDIM;
    for (int i = t; i < 128 * CDIM; i += 256) tile[i] = base[i];
    __syncthreads();
    for (int n = 0; n < 128; ++n) {
      float sd = tile[n * CDIM + d0];
      g0 += tile[n * CDIM + c0]      * sd;
      g1 += tile[n * CDIM + c0 + 8]  * sd;
      g2 += tile[n * CDIM + c0 + 16] * sd;
      g3 += tile[n * CDIM + c0 + 24] * sd;
    }
    __syncthreads();
  }
  float gs = g0 * g0 + g1 * g1 + g2 * g2 + g3 * g3;
  for (int off = 16; off; off >>= 1) gs += __shfl_down(gs, off);
  __shared__ float wsum[8];
  if ((t & 31) == 0) wsum[t >> 5] = gs;
  __syncthreads();
  if (t == 0) {
    float tot = 0;
    for (int i = 0; i < 8; ++i) tot += wsum[i];
    atomicAdd(&acc[3], tot);
  }
}

// ---------------- graph embedding: (1/C) * sum_n Z[b,n,:] -------------------
__global__ void k_gsum(const float* __restrict__ z, float* __restrict__ gsum) {
  int b = blockIdx.x, f = threadIdx.x;  // 64 threads
  const float* base = z + (long long)b * NPG * HIDD + f;
  float s = 0.f;
  for (int n = 0; n < NPG; ++n) s += base[n * HIDD];
  gsum[b * HIDD + f] = s * (1.0f / CDIM);
}

// ---------------- classifier ------------------------------------------------
__global__ void k_cls(const float* __restrict__ gsum, const float* __restrict__ Wc1,
                      const float* __restrict__ bc1, const float* __restrict__ Wc2,
                      const float* __restrict__ bc2, float* __restrict__ out) {
  int b = threadIdx.x;
  if (b >= NBATCH) return;
  const float* g = gsum + b * HIDD;
  float hid[HIDD / 2];
  for (int j = 0; j < HIDD / 2; ++j) {
    float s = bc1[j];
    for (int f = 0; f < HIDD; ++f) s += g[f] * Wc1[f * (HIDD / 2) + j];
    hid[j] = fmaxf(s, 0.f);
  }
  for (int k = 0; k < NCLS; ++k) {
    float s = bc2[k];
    for (int j = 0; j < HIDD / 2; ++j) s += hid[j] * Wc2[j * NCLS + k];
    out[b * NCLS + k] = s;
  }
}

// ---------------- finalize losses ------------------------------------------
__global__ void k_final(const float* __restrict__ acc, float* __restrict__ out) {
  if (threadIdx.x == 0 && blockIdx.x == 0) {
    float sq = acc[0] - 2.f * acc[1] + acc[3];
    sq = fmaxf(sq, 0.f);
    out[NBATCH * NCLS]     = sqrtf(sq) / 67108864.0f;  // /(B*N*N)
    out[NBATCH * NCLS + 1] = acc[2] / 65536.0f;        // /(B*N)
  }
}

// ---------------- host ------------------------------------------------------
extern "C" void kernel_launch(void* const* d_in, const int* in_sizes, int n_in,
                              void* d_out, int out_size, void* d_ws, size_t ws_size,
                              hipStream_t stream) {
  const int*   xidx = (const int*)d_in[0];
  const int*   eidx = (const int*)d_in[1];
  const float* emb  = (const float*)d_in[2];
  const float* Wg   = (const float*)d_in[3];
  const float* bg   = (const float*)d_in[4];
  const float* Wa1  = (const float*)d_in[5];
  const float* ba1  = (const float*)d_in[6];
  const float* Wa2  = (const float*)d_in[7];
  const float* ba2  = (const float*)d_in[8];
  const float* Wc1  = (const float*)d_in[9];
  const float* bc1  = (const float*)d_in[10];
  const float* Wc2  = (const float*)d_in[11];
  const float* bc2  = (const float*)d_in[12];
  float* out = (float*)d_out;

  int E = in_sizes[1] / 2;
  const int* src = eidx;
  const int* dst = eidx + E;

  // workspace layout
  const size_t OFF_H    = 0;                         // 16 MiB
  const size_t OFF_Z    = (size_t)16 << 20;          // 16 MiB
  const size_t OFF_S    = (size_t)32 << 20;          // 8 MiB
  const size_t OFF_DEG  = (size_t)40 << 20;          // 256 KiB
  const size_t OFF_DINV = OFF_DEG + 262144;          // 256 KiB
  const size_t OFF_G    = OFF_DINV + 262144;         // 16 KiB
  const size_t OFF_ACC  = OFF_G + 16384;             // 64 B
  const size_t OFF_A    = (size_t)41 << 20;          // 64 MiB packed counts
  const size_t NEED_A   = OFF_A + ((size_t)64 << 20);

  char* ws = (char*)d_ws;
  float*        h    = (float*)(ws + OFF_H);
  float*        z    = (float*)(ws + OFF_Z);
  float*        Smat = (float*)(ws + OFF_S);
  float*        deg  = (float*)(ws + OFF_DEG);
  float*        dinv = (float*)(ws + OFF_DINV);
  float*        gsum = (float*)(ws + OFF_G);
  float*        acc  = (float*)(ws + OFF_ACC);
  unsigned int* Apk  = (unsigned int*)(ws + OFF_A);
  int use_a = (ws_size >= NEED_A) ? 1 : 0;

  hipMemsetAsync(z, 0, (size_t)TOTAL * HIDD * sizeof(float), stream);
  hipMemsetAsync(acc, 0, 64, stream);
  if (use_a) hipMemsetAsync(Apk, 0, (size_t)64 << 20, stream);

  k_init_deg<<<(TOTAL + 255) / 256, 256, 0, stream>>>(deg);
  k_deg<<<(E + 255) / 256, 256, 0, stream>>>(dst, E, deg);
  k_dinv<<<(TOTAL + 255) / 256, 256, 0, stream>>>(deg, dinv);

  k_embed_gemm<<<TOTAL / 64, 128, 0, stream>>>(xidx, emb, Wg, h);

  k_scatter<<<(E * 32 + 255) / 256, 256, 0, stream>>>(src, dst, E, dinv, h, z);
  k_finish_z<<<(TOTAL * HIDD) / 256, 256, 0, stream>>>(z, h, dinv, bg);

  k_assign<<<TOTAL / 64, 128, 0, stream>>>(z, Wa1, ba1, Wa2, ba2, Smat, acc);

  k_edge_loss<<<(E * 32 + 255) / 256, 256, 0, stream>>>(src, dst, E, Smat, Apk, use_a, acc);
  k_gram<<<NBATCH, 256, 0, stream>>>(Smat, acc);

  k_gsum<<<NBATCH, 64, 0, stream>>>(z, gsum);
  k_cls<<<1, 64, 0, stream>>>(gsum, Wc1, bc1, Wc2, bc2, out);
  k_final<<<1, 32, 0, stream>>>(acc, out);
}